// GIN_24146306138665
// MI455X (gfx1250) — compile-verified
//
#include <hip/hip_runtime.h>

#define N_NODES 100000
#define N_EDGES 1200000
#define ATOM_D  32
#define BOND_D  16
#define H       64
#define OUT_D   128
#define NLAYERS 4
#define LN_EPS  1e-5f

typedef float v2f __attribute__((ext_vector_type(2)));
typedef float v4f __attribute__((ext_vector_type(4)));
typedef float v8f __attribute__((ext_vector_type(8)));

__device__ __forceinline__ float silu_f(float x) {
  return x * (1.0f / (1.0f + __expf(-x)));
}

// ---------------------------------------------------------------------------
// D = act( (A [+ A2]) @ B + bias )
// A: [M, K] row-major, B: [K, NCOLS] row-major, D: [M, NCOLS] row-major.
// One wave32 computes one 16-row stripe of D across all NCOLS/16 column tiles
// using V_WMMA_F32_16X16X4_F32 (f32 in / f32 out — reference precision).
// A operand (16x4 f32): lane&15 = M row, VGPR0/1 = K {kb, kb+1}, kb = 2*(lane>>4).
// B operand (4x16 f32): lane&15 = N col, same K split.
// C/D (16x16 f32): row = v + 8*(lane>>4), col = lane&15.
// ACT: 0 = bias only, 1 = bias + silu.
// A_NT: non-temporal A loads (one-shot stream, don't pollute L2).
// D_NT: non-temporal D stores (stream larger than L2, written once).
// ---------------------------------------------------------------------------
template<int ACT, bool HAS_A2, int K, int NCOLS, bool A_NT = false, bool D_NT = false>
__global__ void gemm_wmma_f32(const float* __restrict__ A,
                              const float* __restrict__ A2,
                              const float* __restrict__ B,
                              const float* __restrict__ bias,
                              float* __restrict__ D,
                              int M) {
  constexpr int NT = NCOLS / 16;
  const int lane = threadIdx.x & 31;
  const int wave = threadIdx.x >> 5;
  const int mt = blockIdx.x * (blockDim.x >> 5) + wave;
  if (mt * 16 >= M) return;

  const int mrow = mt * 16 + (lane & 15);
  const int kb   = (lane >> 4) << 1;   // 0 or 2
  const int ncol = lane & 15;

  v8f acc[NT] = {};

  #pragma unroll
  for (int k0 = 0; k0 < K; k0 += 4) {
    const float* ap = A + (size_t)mrow * K + (k0 + kb);
    v2f a;
    if constexpr (A_NT) {
      a.x = __builtin_nontemporal_load(ap);
      a.y = __builtin_nontemporal_load(ap + 1);
    } else {
      a.x = ap[0];
      a.y = ap[1];
    }
    if constexpr (HAS_A2) {
      const float* ap2 = A2 + (size_t)mrow * K + (k0 + kb);
      a.x += ap2[0];
      a.y += ap2[1];
    }
    const float* bp0 = B + (size_t)(k0 + kb)     * NCOLS + ncol;
    const float* bp1 = B + (size_t)(k0 + kb + 1) * NCOLS + ncol;
    #pragma unroll
    for (int t = 0; t < NT; ++t) {
      v2f b;
      b.x = bp0[t * 16];
      b.y = bp1[t * 16];
      acc[t] = __builtin_amdgcn_wmma_f32_16x16x4_f32(
          false, a, false, b, (short)0, acc[t], false, false);
    }
  }

  const int rbase = mt * 16 + ((lane >> 4) << 3);
  #pragma unroll
  for (int t = 0; t < NT; ++t) {
    const int c = t * 16 + ncol;
    const float bv = bias[c];
    #pragma unroll
    for (int v = 0; v < 8; ++v) {
      float o = acc[t][v] + bv;
      if (ACT == 1) o = silu_f(o);
      float* dp = D + (size_t)(rbase + v) * NCOLS + c;
      if constexpr (D_NT) {
        __builtin_nontemporal_store(o, dp);
      } else {
        *dp = o;
      }
    }
  }
}

// ---------------------------------------------------------------------------
// Edge message + scatter: agg[dst[e]] += relu(h[src[e]] + ea[e])
// 16 threads per edge, 4 columns (16B) per thread.
// ea (307 MB, > L2) is streamed with non-temporal loads so that h/agg
// (25.6 MB each) stay resident in the 192 MB L2 for the gather + atomics.
// Atomics: hardware global_atomic_add_f32 (relaxed, agent scope).
// ---------------------------------------------------------------------------
__global__ void edge_scatter(const float* __restrict__ h,
                             const float* __restrict__ ea,
                             const int* __restrict__ src,
                             const int* __restrict__ dst,
                             float* __restrict__ agg) {
  const long long gid = (long long)blockIdx.x * blockDim.x + threadIdx.x;
  const int e = (int)(gid >> 4);
  if (e >= N_EDGES) return;
  const int c = (int)(gid & 15) << 2;

  const int s = src[e];
  const int d = dst[e];

  const v4f hv = *(const v4f*)(h + (size_t)s * H + c);
  const v4f ev =
      __builtin_nontemporal_load((const v4f*)(ea + (size_t)e * H + c));

  float m0 = fmaxf(hv.x + ev.x, 0.0f);
  float m1 = fmaxf(hv.y + ev.y, 0.0f);
  float m2 = fmaxf(hv.z + ev.z, 0.0f);
  float m3 = fmaxf(hv.w + ev.w, 0.0f);

  float* out = agg + (size_t)d * H + c;
  __hip_atomic_fetch_add(out + 0, m0, __ATOMIC_RELAXED, __HIP_MEMORY_SCOPE_AGENT);
  __hip_atomic_fetch_add(out + 1, m1, __ATOMIC_RELAXED, __HIP_MEMORY_SCOPE_AGENT);
  __hip_atomic_fetch_add(out + 2, m2, __ATOMIC_RELAXED, __HIP_MEMORY_SCOPE_AGENT);
  __hip_atomic_fetch_add(out + 3, m3, __ATOMIC_RELAXED, __HIP_MEMORY_SCOPE_AGENT);
}

// ---------------------------------------------------------------------------
// In-place LayerNorm(+affine) + silu over rows of length H=64.
// One wave32 per row: 2 elements/lane, shfl_xor tree reduction.
// ---------------------------------------------------------------------------
__global__ void ln_silu_kernel(float* __restrict__ t,
                               const float* __restrict__ g,
                               const float* __restrict__ be,
                               int M) {
  const int lane = threadIdx.x & 31;
  const int row = blockIdx.x * (blockDim.x >> 5) + (threadIdx.x >> 5);
  if (row >= M) return;

  float* p = t + (size_t)row * H;
  float v0 = p[lane];
  float v1 = p[lane + 32];
  float sum = v0 + v1;
  float sq  = v0 * v0 + v1 * v1;
  #pragma unroll
  for (int off = 16; off >= 1; off >>= 1) {
    sum += __shfl_xor(sum, off, 32);
    sq  += __shfl_xor(sq, off, 32);
  }
  const float mean = sum * (1.0f / H);
  const float var  = sq * (1.0f / H) - mean * mean;
  const float rstd = rsqrtf(var + LN_EPS);

  float o0 = (v0 - mean) * rstd * g[lane]      + be[lane];
  float o1 = (v1 - mean) * rstd * g[lane + 32] + be[lane + 32];
  p[lane]      = silu_f(o0);
  p[lane + 32] = silu_f(o1);
}

// ---------------------------------------------------------------------------
extern "C" void kernel_launch(void* const* d_in, const int* in_sizes, int n_in,
                              void* d_out, int out_size, void* d_ws, size_t ws_size,
                              hipStream_t stream) {
  const float* x         = (const float*)d_in[0];
  const float* edge_attr = (const float*)d_in[1];
  const int*   edge_idx  = (const int*)  d_in[2];
  // d_in[3] = batch (unused by the reference computation)
  const float* Wa  = (const float*)d_in[4];
  const float* ba  = (const float*)d_in[5];
  const float* Wb  = (const float*)d_in[6];
  const float* bb  = (const float*)d_in[7];
  const float* W1  = (const float*)d_in[8];
  const float* b1  = (const float*)d_in[9];
  const float* g1  = (const float*)d_in[10];
  const float* be1 = (const float*)d_in[11];
  const float* W2  = (const float*)d_in[12];
  const float* b2  = (const float*)d_in[13];
  const float* Wl  = (const float*)d_in[14];
  const float* bl  = (const float*)d_in[15];
  float* out = (float*)d_out;

  // Workspace layout (all fully written before read each call):
  float* h    = (float*)d_ws;                    // [N, H]
  float* agg  = h    + (size_t)N_NODES * H;      // [N, H]
  float* tbuf = agg  + (size_t)N_NODES * H;      // [N, H]
  float* ea   = tbuf + (size_t)N_NODES * H;      // [E, H]

  const int* src = edge_idx;             // edge_index[0]
  const int* dst = edge_idx + N_EDGES;   // edge_index[1]

  const int WPB = 8;            // waves per block
  const int TPB = WPB * 32;     // 256 threads

  const int node_tiles = N_NODES / 16;  // 6250 (exact)
  const int edge_tiles = N_EDGES / 16;  // 75000 (exact)
  const int node_blocks = (node_tiles + WPB - 1) / WPB;
  const int edge_blocks = (edge_tiles + WPB - 1) / WPB;

  // Encoders: h = x @ Wa + ba ; ea = edge_attr @ Wb + bb
  gemm_wmma_f32<0, false, ATOM_D, H><<<node_blocks, TPB, 0, stream>>>(
      x, nullptr, Wa, ba, h, N_NODES);
  // ea is a one-shot 307 MB stream (> L2): NT loads for edge_attr, NT stores for ea.
  gemm_wmma_f32<0, false, BOND_D, H, true, true><<<edge_blocks, TPB, 0, stream>>>(
      edge_attr, nullptr, Wb, bb, ea, N_EDGES);

  const long long scat_threads = (long long)N_EDGES * 16;
  const int scat_blocks = (int)((scat_threads + 255) / 256);
  const int ln_blocks = (N_NODES + WPB - 1) / WPB;

  for (int l = 0; l < NLAYERS; ++l) {
    (void)hipMemsetAsync(agg, 0, (size_t)N_NODES * H * sizeof(float), stream);
    edge_scatter<<<scat_blocks, 256, 0, stream>>>(h, ea, src, dst, agg);

    // tbuf = (h + agg) @ W1[l] + b1[l]
    gemm_wmma_f32<0, true, H, H><<<node_blocks, TPB, 0, stream>>>(
        h, agg, W1 + (size_t)l * H * H, b1 + (size_t)l * H, tbuf, N_NODES);

    // tbuf = silu(LN(tbuf) * g1[l] + be1[l])
    ln_silu_kernel<<<ln_blocks, TPB, 0, stream>>>(
        tbuf, g1 + (size_t)l * H, be1 + (size_t)l * H, N_NODES);

    // h = silu(tbuf @ W2[l] + b2[l])
    gemm_wmma_f32<1, false, H, H><<<node_blocks, TPB, 0, stream>>>(
        tbuf, nullptr, W2 + (size_t)l * H * H, b2 + (size_t)l * H, h, N_NODES);
  }

  // out = silu(h @ Wl + bl)
  gemm_wmma_f32<1, false, H, OUT_D><<<node_blocks, TPB, 0, stream>>>(
      h, nullptr, Wl, bl, out, N_NODES);
}